// CtcLoss_83288005804678
// MI455X (gfx1250) — compile-verified
//
#include <hip/hip_runtime.h>

// Problem constants (fixed by the reference).
constexpr int Bc = 512;   // batch
constexpr int Tc = 256;   // time steps
constexpr int Cc = 512;   // classes (class 0 = blank)
constexpr int Sc = 32;    // max target length
constexpr int Lc = 2 * Sc + 1;  // 65 extended states
#define NEG_INF (-1e30f)

__device__ __forceinline__ float lse2(float a, float b) {
  const float m = fmaxf(a, b);
  return m + logf(expf(a - m) + expf(b - m));
}
__device__ __forceinline__ float lse3(float a, float b, float c) {
  const float m = fmaxf(a, fmaxf(b, c));
  return m + logf(expf(a - m) + expf(b - m) + expf(c - m));
}

// ---------------------------------------------------------------------------
// Kernel 1: per-(b,t) logsumexp over C + gather of 33 needed classes.
// One wave32 per row, 8 waves/block, 8 rows per wave with an async
// global->LDS double-buffered pipeline (CDNA5 ASYNCcnt path). pred is
// read once -> non-temporal hint keeps the lp workspace resident in L2.
// Output: lp[row*33 + 0] = blank logprob, lp[row*33 + 1 + j] = label j.
// ---------------------------------------------------------------------------
#define ROWS_PER_WAVE 8

__global__ __launch_bounds__(256) void k_rowlse(const float* __restrict__ pred,
                                                const int* __restrict__ target,
                                                float* __restrict__ lp) {
  __shared__ __align__(16) float lbuf[8][2][Cc];  // 8 waves x 2 buffers x 2KB

  const int wave = threadIdx.x >> 5;
  const int lane = threadIdx.x & 31;
  const long gwave = (long)blockIdx.x * 8 + wave;
  const long row0 = gwave * ROWS_PER_WAVE;

  // All 8 rows of this wave share one batch element (8 divides Tc=256).
  const int b = (int)(row0 >> 8);
  const int tj = target[b * Sc + lane];  // labels in [1, C-1]

  // LDS byte offsets (low 32 bits of the generic pointer = LDS address).
  unsigned ldsBase[2];
  ldsBase[0] = (unsigned)(size_t)(const void*)&lbuf[wave][0][0];
  ldsBase[1] = (unsigned)(size_t)(const void*)&lbuf[wave][1][0];

  // Issue one row (2KB) as 4 async b128-per-lane copies into buffer `buf`.
  auto issue_row = [&](int buf, long row) {
    const float* gp = pred + row * Cc + lane * 4;
    unsigned l0 = ldsBase[buf] + (unsigned)(lane * 16);
#pragma unroll
    for (int k = 0; k < 4; ++k) {
      asm volatile("global_load_async_to_lds_b128 %0, %1, off th:TH_LOAD_NT"
                   :
                   : "v"(l0 + (unsigned)(k * 512)), "v"(gp + k * 128)
                   : "memory");
    }
  };

  int buf = 0;
  issue_row(0, row0);

  for (int i = 0; i < ROWS_PER_WAVE; ++i) {
    const long row = row0 + i;
    if (i + 1 < ROWS_PER_WAVE) {
      issue_row(buf ^ 1, row + 1);
      // 8 outstanding, wait until <=4 -> current buffer complete (in-order).
      asm volatile("s_wait_asynccnt 0x4" ::: "memory");
    } else {
      asm volatile("s_wait_asynccnt 0x0" ::: "memory");
    }

    const float* rb = &lbuf[wave][buf][0];
    const float4 v0 = *(const float4*)(rb + lane * 4);
    const float4 v1 = *(const float4*)(rb + lane * 4 + 128);
    const float4 v2 = *(const float4*)(rb + lane * 4 + 256);
    const float4 v3 = *(const float4*)(rb + lane * 4 + 384);

    // Row max: 16 in-lane values, then 5-step wave32 butterfly.
    float m = fmaxf(fmaxf(fmaxf(v0.x, v0.y), fmaxf(v0.z, v0.w)),
                    fmaxf(fmaxf(v1.x, v1.y), fmaxf(v1.z, v1.w)));
    m = fmaxf(m, fmaxf(fmaxf(fmaxf(v2.x, v2.y), fmaxf(v2.z, v2.w)),
                       fmaxf(fmaxf(v3.x, v3.y), fmaxf(v3.z, v3.w))));
#pragma unroll
    for (int off = 16; off >= 1; off >>= 1)
      m = fmaxf(m, __shfl_xor(m, off, 32));

    // Sum of exp(v - m).
    float s = expf(v0.x - m) + expf(v0.y - m) + expf(v0.z - m) + expf(v0.w - m);
    s += expf(v1.x - m) + expf(v1.y - m) + expf(v1.z - m) + expf(v1.w - m);
    s += expf(v2.x - m) + expf(v2.y - m) + expf(v2.z - m) + expf(v2.w - m);
    s += expf(v3.x - m) + expf(v3.y - m) + expf(v3.z - m) + expf(v3.w - m);
#pragma unroll
    for (int off = 16; off >= 1; off >>= 1)
      s += __shfl_xor(s, off, 32);

    const float lse = m + logf(s);

    // Gather the 33 needed classes straight from LDS.
    float* lpr = lp + row * 33;
    lpr[1 + lane] = rb[tj] - lse;
    if (lane == 0) lpr[0] = rb[0] - lse;

    buf ^= 1;
  }
}

// ---------------------------------------------------------------------------
// Kernel 2: CTC forward recurrence. One wave32 per batch element; lane l
// holds extended states s=2l and s=2l+1 (lane 31 also s=64). The only
// cross-lane dependency is prev lane's odd-state alpha -> one shfl_up per
// step, zero barriers. Emissions software-pipelined 16 steps deep so L2
// latency is hidden despite ~2 waves/WGP occupancy.
// ---------------------------------------------------------------------------
__global__ __launch_bounds__(32) void k_ctc(const float* __restrict__ lp,
                                            const int* __restrict__ target,
                                            const int* __restrict__ length,
                                            float* __restrict__ lossBuf) {
  __shared__ float fin[Lc];
  const int b = blockIdx.x;
  const int lane = threadIdx.x;  // 0..31
  const float* lpb = lp + (long)b * Tc * 33;

  // skip flag for odd state s=2*lane+1 (label index j = lane):
  // allowed iff j>=1 and target[j] != target[j-1].
  bool skip = false;
  if (lane >= 1) skip = (target[b * Sc + lane] != target[b * Sc + lane - 1]);

  // alpha at t=0: only states 0 (blank) and 1 (first label) are live.
  float a0 = (lane == 0) ? lpb[0] : NEG_INF;  // state 2*lane
  float a1 = (lane == 0) ? lpb[1] : NEG_INF;  // state 2*lane+1
  float a2 = NEG_INF;                         // state 64 (lane 31 only)

  constexpr int PF = 16;  // emission prefetch depth (register pipeline)
  float pB[PF], pL[PF];
#pragma unroll
  for (int d = 0; d < PF; ++d) {
    const int tt = 1 + d;
    pB[d] = lpb[tt * 33];             // blank emission (uniform)
    pL[d] = lpb[tt * 33 + 1 + lane];  // label emission for this lane
  }

  for (int g = 0; g < 16; ++g) {
#pragma unroll
    for (int u = 0; u < PF; ++u) {
      const int t = 1 + g * PF + u;
      if (t < Tc) {
        const float eB = pB[u];
        const float eL = pL[u];
        const int tn = t + PF;
        if (tn < Tc) {
          pB[u] = lpb[tn * 33];
          pL[u] = lpb[tn * 33 + 1 + lane];
        }
        float p1 = __shfl_up(a1, 1, 32);        // alpha[2*lane-1] (old)
        if (lane == 0) p1 = NEG_INF;
        const float n0 = lse2(a0, p1) + eB;                       // even state
        const float n1 = lse3(a1, a0, skip ? p1 : NEG_INF) + eL;  // odd state
        const float n2 = lse2(a2, a1) + eB;                       // state 64
        a0 = n0;
        a1 = n1;
        a2 = (lane == 31) ? n2 : NEG_INF;
      }
    }
  }

  fin[2 * lane] = a0;
  fin[2 * lane + 1] = a1;
  if (lane == 31) fin[64] = a2;
  __syncthreads();

  if (lane == 0) {
    const int len = length[b];
    const int end = 2 * len;  // trailing blank index in extended sequence
    float loss = -lse2(fin[end], fin[end - 1]);
    if (loss >= 1e29f) loss = 0.0f;  // zero_infinity
    lossBuf[b] = loss / (float)((len > 0) ? len : 1);
  }
}

// ---------------------------------------------------------------------------
// Kernel 3: mean over batch.
// ---------------------------------------------------------------------------
__global__ __launch_bounds__(256) void k_mean(const float* __restrict__ lossBuf,
                                              float* __restrict__ out) {
  __shared__ float red[256];
  red[threadIdx.x] = lossBuf[threadIdx.x] + lossBuf[threadIdx.x + 256];
  __syncthreads();
#pragma unroll
  for (int o = 128; o > 0; o >>= 1) {
    if (threadIdx.x < o) red[threadIdx.x] += red[threadIdx.x + o];
    __syncthreads();
  }
  if (threadIdx.x == 0) out[0] = red[0] / (float)Bc;
}

// ---------------------------------------------------------------------------
extern "C" void kernel_launch(void* const* d_in, const int* in_sizes, int n_in,
                              void* d_out, int out_size, void* d_ws,
                              size_t ws_size, hipStream_t stream) {
  const float* pred = (const float*)d_in[0];  // [B,T,C] f32
  const int* target = (const int*)d_in[1];    // [B,S] i32
  const int* length = (const int*)d_in[2];    // [B] i32
  (void)in_sizes; (void)n_in; (void)out_size; (void)ws_size;

  float* lp = (float*)d_ws;                    // B*T*33 floats (~17.3 MB)
  float* lossBuf = lp + (size_t)Bc * Tc * 33;  // B floats

  // Phase 1: 131072 rows, 1 wave/row, 8 rows/wave, 8 waves/block.
  k_rowlse<<<dim3(2048), dim3(256), 0, stream>>>(pred, target, lp);
  // Phase 2: forward recurrence, one wave32 per batch element.
  k_ctc<<<dim3(Bc), dim3(32), 0, stream>>>(lp, target, length, lossBuf);
  // Phase 3: mean.
  k_mean<<<dim3(1), dim3(256), 0, stream>>>(lossBuf, (float*)d_out);
}